// WignerKernelReducedCost_49349174231047
// MI455X (gfx1250) — compile-verified
//
#include <hip/hip_runtime.h>

// ---------------------------------------------------------------------------
// WignerKernelReducedCost for MI455X (gfx1250, wave32).
// Compute-bound (~24 GFLOP fp32 vs ~3us of HBM traffic at 23.3 TB/s):
//  * combine(X,X,3) is symmetric in (l1,l2) because both operands are the
//    same input (CG symmetry sign cancels squared) -> only l1<=l2 combos are
//    computed, weight 2 for l1<l2: ~30% less work in every stage.
//  * Stage-1 of each combo runs on V_WMMA_F32_16X16X4_F32: pre-baked
//    zero-padded CG fragments (one ds_load_b64) x 16 batch columns.
//  * Stages 2/3 + nu=3/4 reductions on VALU (co-executes with XDL WMMA).
//  * All staging through LDS (~296KB of the 320KB WGP budget).
// ---------------------------------------------------------------------------

#define NCOMBO 23
#define CG_TOTAL 2351
#define NFRAGS 76
#define FRAG_TOTAL (NFRAGS * 64)
#define TAB_TOTAL (CG_TOTAL + FRAG_TOTAL)
#define WAVES_PER_BLOCK 4
#define TB (WAVES_PER_BLOCK * 32)
#define ZROW 84   /* dedicated zero row in the transposed input tile */

typedef float v2f __attribute__((ext_vector_type(2)));
typedef float v8f __attribute__((ext_vector_type(8)));

// Combos (l1<=l2 only), L in |l1-l2|..min(l1+l2,3); weight 2 when l1<l2.
__constant__ int c_l1[NCOMBO] = {0,0,0,0, 1,1,1,1,1,1,1,1, 2,2,2,2,2,2,2, 3,3,3,3};
__constant__ int c_l2[NCOMBO] = {0,1,2,3, 1,1,1,2,2,2,3,3, 2,2,2,2,3,3,3, 3,3,3,3};
__constant__ int c_L [NCOMBO] = {0,1,2,3, 0,1,2,1,2,3,2,3, 0,1,2,3,1,2,3, 0,1,2,3};
__constant__ float c_w[NCOMBO] = {1,2,2,2, 1,1,1,2,2,2,2,2, 1,1,1,1,2,2,2, 1,1,1,1};
// Flat offsets of each dense C[l1,l2,L] table ([m][n][k], size n1*n2*nL).
__constant__ int c_off[NCOMBO] = {0,1,10,35, 84,93,120,165,210,285,390,495,
                                  642,667,742,867,1042,1147,1322, 1567,1616,1763,2008};
// Fragment-table base (units of 64-float fragments), order [mt][kc] per combo.
__constant__ int c_foff[NCOMBO] = {0,1,2,4, 8,9,10,11,12,14,17,20,
                                   24,26,28,32,38,42,48, 56,58,62,68};
// Offsets of the x_l blocks inside the concatenated 84-float element vector.
__constant__ int d_xoff[4] = {0, 1, 10, 35};
// Row bases of X2 keys 'L_+1' / 'L_-1' inside the 167-row X2 accumulator.
__constant__ int d_x2pos[4] = {0, 1, 19, 69};
__constant__ int d_x2neg[4] = {0, 10, 44, 118};   // L=0 negative never occurs
// Combo indices of the diagonal (l,l,0) tables used by the nu=3/4 reductions.
__constant__ int d_diag[4] = {0, 4, 12, 19};

// --- Kernel 1: CG tables (Racah) + zero-padded WMMA A-fragments in d_ws ----
__global__ void wigner_cg_init(float* __restrict__ ws) {
  double f[16];
  f[0] = 1.0;
  for (int i = 1; i < 16; ++i) f[i] = f[i - 1] * i;
  for (int c = 0; c < NCOMBO; ++c) {
    const int l1 = c_l1[c], l2 = c_l2[c], L = c_L[c];
    const int n2 = 2 * l2 + 1, nL = 2 * L + 1;
    const int sz = (2 * l1 + 1) * n2 * nL, off = c_off[c];
    const double pref0 =
        sqrt((2.0 * L + 1.0) * f[L + l1 - l2] * f[L - l1 + l2] * f[l1 + l2 - L] / f[l1 + l2 + L + 1]);
    for (int t = threadIdx.x; t < sz; t += blockDim.x) {
      const int m = t / (n2 * nL), rem = t % (n2 * nL), n = rem / nL, k = rem % nL;
      const int m1 = m - l1, m2 = n - l2, M = m1 + m2;
      double val = 0.0;
      if (M >= -L && M <= L && k == M + L) {
        const double pref =
            pref0 * sqrt(f[L + M] * f[L - M] * f[l1 - m1] * f[l1 + m1] * f[l2 - m2] * f[l2 + m2]);
        double s = 0.0;
        for (int kk = 0; kk <= l1 + l2 - L; ++kk) {
          const int a1 = l1 - m1 - kk, a2 = l2 + m2 - kk;
          const int a3 = L - l2 + m1 + kk, a4 = L - l1 - m2 + kk;
          if (a1 < 0 || a2 < 0 || a3 < 0 || a4 < 0) continue;
          const double term = 1.0 / (f[kk] * f[l1 + l2 - L - kk] * f[a1] * f[a2] * f[a3] * f[a4]);
          s += (kk & 1) ? -term : term;
        }
        val = pref * s;
      }
      ws[off + t] = (float)val;
    }
  }
  __syncthreads();
  // A-fragment bake: frag (mt,kc): value(lane,j) = C[m,row] with zero padding,
  // row = mt*16 + (lane&15), m = kc*4 + (lane>=16 ? 2 : 0) + j.
  for (int c = 0; c < NCOMBO; ++c) {
    const int l1 = c_l1[c], l2 = c_l2[c], L = c_L[c];
    const int n1 = 2 * l1 + 1, R = (2 * l2 + 1) * (2 * L + 1);
    const int cgo = c_off[c], fo = c_foff[c];
    const int mtiles = (R + 15) >> 4, kch = (n1 + 3) >> 2;
    for (int t = threadIdx.x; t < mtiles * kch * 64; t += blockDim.x) {
      const int fi = t >> 6, s = t & 63;
      const int mt = fi / kch, kc = fi % kch;
      const int ln = s >> 1, j = s & 1;
      const int row = mt * 16 + (ln & 15);
      const int m = kc * 4 + ((ln >> 4) ? 2 : 0) + j;
      ws[CG_TOTAL + (fo + fi) * 64 + s] =
          (row < R && m < n1) ? ws[cgo + m * R + row] : 0.f;
    }
  }
}

// --- Kernel 2: main Wigner contraction -------------------------------------
__launch_bounds__(TB, 1)
__global__ void wigner_main(const float* __restrict__ x0, const float* __restrict__ x1,
                            const float* __restrict__ x2, const float* __restrict__ x3,
                            const float* __restrict__ tab, float* __restrict__ out, int Btot) {
  __shared__ __align__(16) float sTAB[TAB_TOTAL];       // CG tables + A-frags
  __shared__ float sXT[WAVES_PER_BLOCK][85 * 16];        // inputs [comp][b], row 84 = zeros
  __shared__ float sA [WAVES_PER_BLOCK][64 * 7 * 16];    // stage-1, rows padded to 16
  __shared__ float sE [WAVES_PER_BLOCK][7 * 7 * 7 * 16]; // stage-2 [(k,q,p)][b]
  __shared__ float sX2[WAVES_PER_BLOCK][167 * 16];       // X2 accumulator [row][b]

  const int tid = threadIdx.x, lane = tid & 31, wid = tid >> 5;
  const int b = lane & 15, half = lane >> 4;
  const float* sCG = sTAB;
  const float* sFRAG = sTAB + CG_TOTAL;

  for (int t = tid; t < TAB_TOTAL; t += TB) sTAB[t] = tab[t];

  const int tile = blockIdx.x * WAVES_PER_BLOCK + wid;
  const int base = tile * 16;
  float* XT = sXT[wid];
  float* A  = sA[wid];
  float* E  = sE[wid];
  float* X2 = sX2[wid];

  // Prefetch next block's largest input into L2 (global_prefetch_b8).
  {
    int pe = base + WAVES_PER_BLOCK * 16;
    if (pe < Btot) __builtin_prefetch(&x3[(size_t)pe * 49], 0, 1);
  }

  // Gather the 16-element tile transposed into LDS (clamped tail indices).
  for (int t = lane; t < 84 * 16; t += 32) {
    const int bb = t / 84, comp = t % 84;
    int e = base + bb;
    e = (e < Btot) ? e : (Btot - 1);
    float v;
    if (comp < 1)       v = x0[e];
    else if (comp < 10) v = x1[(size_t)e * 9 + (comp - 1)];
    else if (comp < 35) v = x2[(size_t)e * 25 + (comp - 10)];
    else                v = x3[(size_t)e * 49 + (comp - 35)];
    XT[comp * 16 + bb] = v;
  }
  for (int t = lane; t < 16; t += 32) XT[ZROW * 16 + t] = 0.f;   // zero row
  for (int t = lane; t < 167 * 16; t += 32) X2[t] = 0.f;
  __syncthreads();

  for (int c = 0; c < NCOMBO; ++c) {
    const int l1 = c_l1[c], l2 = c_l2[c], L = c_L[c];
    const int n1 = 2 * l1 + 1, n2 = 2 * l2 + 1, nL = 2 * L + 1;
    const int R = n2 * nL, cgo = c_off[c], fb = c_foff[c];
    const int xo1 = d_xoff[l1], xo2 = d_xoff[l2];
    const int mtiles = (R + 15) >> 4, kch = (n1 + 3) >> 2;

    // ---- Stage 1 (WMMA): A[b,(n,k),p] = sum_m C[m,(n,k)] * X_l1[b,m,p] ----
    // A operand: pre-baked CG fragment (one ds_load_b64). B operand: batch
    // columns, padding K rows address-selected to the zero row (no branches).
    // kch is uniform per combo: two separate branch-free inner-loop variants.
    const int m0a = half ? 2 : 0;
    if (kch == 1) {
      for (int p = 0; p < n1; ++p) {
        v2f bf0;
        const int r0 = (m0a     < n1) ? (xo1 + m0a * n1 + p)       : ZROW;
        const int r1 = (m0a + 1 < n1) ? (xo1 + (m0a + 1) * n1 + p) : ZROW;
        bf0.x = XT[r0 * 16 + b];
        bf0.y = XT[r1 * 16 + b];
        for (int mt = 0; mt < mtiles; ++mt) {
          v8f acc = {0.f, 0.f, 0.f, 0.f, 0.f, 0.f, 0.f, 0.f};
          const v2f af0 = *(const v2f*)&sFRAG[(fb + mt) * 64 + lane * 2];
          acc = __builtin_amdgcn_wmma_f32_16x16x4_f32(false, af0, false, bf0,
                                                      (short)0, acc, false, false);
          const int rb = mt * 16 + half * 8;
#pragma unroll
          for (int r = 0; r < 8; ++r)   // rows padded: unconditional stores
            A[((rb + r) * n1 + p) * 16 + b] = acc[r];
        }
      }
    } else {
      for (int p = 0; p < n1; ++p) {
        v2f bf0, bf1;
        {
          const int r0 = xo1 + m0a * n1 + p;             // m0a < n1 (n1 >= 5)
          const int r1 = xo1 + (m0a + 1) * n1 + p;
          bf0.x = XT[r0 * 16 + b];
          bf0.y = XT[r1 * 16 + b];
        }
        {
          const int m0 = 4 + m0a;
          const int r0 = (m0     < n1) ? (xo1 + m0 * n1 + p)       : ZROW;
          const int r1 = (m0 + 1 < n1) ? (xo1 + (m0 + 1) * n1 + p) : ZROW;
          bf1.x = XT[r0 * 16 + b];
          bf1.y = XT[r1 * 16 + b];
        }
        for (int mt = 0; mt < mtiles; ++mt) {
          v8f acc = {0.f, 0.f, 0.f, 0.f, 0.f, 0.f, 0.f, 0.f};
          const float* fp = &sFRAG[(fb + mt * 2) * 64 + lane * 2];
          const v2f af0 = *(const v2f*)fp;
          acc = __builtin_amdgcn_wmma_f32_16x16x4_f32(false, af0, false, bf0,
                                                      (short)0, acc, false, false);
          const v2f af1 = *(const v2f*)(fp + 64);
          acc = __builtin_amdgcn_wmma_f32_16x16x4_f32(false, af1, false, bf1,
                                                      (short)0, acc, false, false);
          const int rb = mt * 16 + half * 8;
#pragma unroll
          for (int r = 0; r < 8; ++r)
            A[((rb + r) * n1 + p) * 16 + b] = acc[r];
        }
      }
    }
    __syncthreads();

    // ---- Stage 2 (VALU): E[b,k,q,p] = sum_n X_l2[b,n,q] * A[b,n,k,p] ------
    for (int q = 0; q < n2; ++q) {
      float xv[7];
#pragma unroll
      for (int n = 0; n < 7; ++n)
        xv[n] = (n < n2) ? XT[(xo2 + n * n2 + q) * 16 + b] : 0.f;
      const int kp = nL * n1;
      for (int idx = half; idx < kp; idx += 2) {
        const int p = idx % n1, k = idx / n1;
        float e = 0.f;
#pragma unroll
        for (int n = 0; n < 7; ++n)
          if (n < n2)
            e += xv[n] * A[((n * nL + k) * n1 + p) * 16 + b];
        E[((k * n2 + q) * n1 + p) * 16 + b] = e;
      }
    }
    __syncthreads();

    // ---- Stage 3 (VALU): contrib[b,k,j] = sum_{p,q} C[p,q,j] * E[b,k,q,p],
    //      accumulated into X2 with symmetry weight (2 when l1<l2). ---------
    const int x2base = ((l1 + l2 + L) & 1) ? d_x2neg[L] : d_x2pos[L];
    const float wc = c_w[c];
    for (int idx = half; idx < nL * nL; idx += 2) {
      const int j = idx % nL, k = idx / nL;
      float s = 0.f;
      for (int pq = 0; pq < n1 * n2; ++pq) {
        const float cc = sCG[cgo + pq * nL + j];
        if (cc != 0.f) {
          const int p = pq / n2, q = pq % n2;
          s += cc * E[((k * n2 + q) * n1 + p) * 16 + b];
        }
      }
      X2[(x2base + idx) * 16 + b] += wc * s;
    }
    __syncthreads();
  }

  // ---- nu outputs ----------------------------------------------------------
  if (lane < 16) {
    const int e = base + b;
    if (e < Btot) {
      const float nu1 = XT[0 * 16 + b];
      const float nu2 = X2[0 * 16 + b];
      float nu3 = 0.f, nu4 = 0.f;
      for (int l = 0; l <= 3; ++l) {
        const int n = 2 * l + 1;
        const int cgo = c_off[d_diag[l]];            // C(l,l,0), flat [m*n+nn]
        const int xo = d_xoff[l];
        const int xp = d_x2pos[l], xn = d_x2neg[l];
        for (int mn = 0; mn < n * n; ++mn) {
          const float c1 = sCG[cgo + mn];
          if (c1 == 0.f) continue;
          const int m = mn / n, nn = mn % n;
          for (int pq = 0; pq < n * n; ++pq) {
            const float c2 = sCG[cgo + pq];
            if (c2 == 0.f) continue;
            const int p = pq / n, q = pq % n;
            const float w = c1 * c2;
            // nu=3: combine(X2,X,0)['0_1'] -> only s=+1 blocks of X2.
            nu3 += w * X2[(xp + m * n + p) * 16 + b] * XT[(xo + nn * n + q) * 16 + b];
            // nu=4: combine(X2,X2,0)['0_1'] -> s1==s2 pairs.
            nu4 += w * X2[(xp + m * n + p) * 16 + b] * X2[(xp + nn * n + q) * 16 + b];
            if (l > 0)
              nu4 += w * X2[(xn + m * n + p) * 16 + b] * X2[(xn + nn * n + q) * 16 + b];
          }
        }
      }
      out[(size_t)e * 4 + 0] = nu1;
      out[(size_t)e * 4 + 1] = nu2;
      out[(size_t)e * 4 + 2] = nu3;
      out[(size_t)e * 4 + 3] = nu4;
    }
  }
}

extern "C" void kernel_launch(void* const* d_in, const int* in_sizes, int n_in,
                              void* d_out, int out_size, void* d_ws, size_t ws_size,
                              hipStream_t stream) {
  const float* x0 = (const float*)d_in[0];
  const float* x1 = (const float*)d_in[1];
  const float* x2 = (const float*)d_in[2];
  const float* x3 = (const float*)d_in[3];
  float* tab = (float*)d_ws;                // needs TAB_TOTAL*4 = 28.9 KB
  const int Btot = in_sizes[0];             // x0 is [B,1,1]

  wigner_cg_init<<<1, 256, 0, stream>>>(tab);

  const int tiles = (Btot + 15) / 16;
  const int blocks = (tiles + WAVES_PER_BLOCK - 1) / WAVES_PER_BLOCK;
  wigner_main<<<blocks, TB, 0, stream>>>(x0, x1, x2, x3, tab, (float*)d_out, Btot);
}